// MicResponseAugment_66185446031578
// MI455X (gfx1250) — compile-verified
//
#include <hip/hip_runtime.h>
#include <cmath>

typedef float v2f __attribute__((ext_vector_type(2)));
typedef float v4f __attribute__((ext_vector_type(4)));
typedef float v8f __attribute__((ext_vector_type(8)));

#define T_LEN   160000
#define N_CH    128
#define SEG     4096
#define WARM    2048
#define NSEG    ((T_LEN + SEG - 1) / SEG)   // 40
#define TSTR    18                           // LDS tile row stride (floats)

struct Coef { float b0, b1, b2, a1, a2; };

// One wave (32 lanes) handles 16 channels for one time segment.
// Per 16-sample block: D(16x16) = H(16x16,lower-tri) x XF(16x16) + state seed,
// realized as 4 chained v_wmma_f32_16x16x4_f32, cascaded for both biquads.
__global__ __launch_bounds__(32)
void MicResponseAugment_biquad_wmma(const float* __restrict__ xin,
                                    float* __restrict__ yout,
                                    Coef hp, Coef lp)
{
    __shared__ float xt [16 * TSTR];   // raw input tile   [ch][t]
    __shared__ float y1t[16 * TSTR];   // highpass output  [ch][t]
    __shared__ float y2t[16 * TSTR];   // lowpass  output  [ch][t]
    __shared__ float hbuf[2][3][16];   // [filter][h|p1|p2][m]

    const int lane  = threadIdx.x;     // 0..31
    const int hi    = lane >> 4;       // lane half (WMMA row/K split)
    const int n     = lane & 15;       // channel column / matrix row m
    const int ch    = lane >> 1;       // staging: channel
    const int halfb = lane & 1;        // staging: which 8 samples

    const int cg      = blockIdx.x & 7;
    const int seg     = blockIdx.x >> 3;
    const int chbase  = cg * 16;
    const int t_start = seg * SEG;
    if (t_start >= T_LEN) return;
    const int t_end = min(t_start + SEG, T_LEN);
    const int warm  = max(t_start - WARM, 0);

    // ---- impulse-response tables (identical in every lane) ----
    {
        #pragma unroll
        for (int f = 0; f < 2; ++f) {
            const float a1 = f ? lp.a1 : hp.a1;
            const float a2 = f ? lp.a2 : hp.a2;
            float h[16], p1[16], p2[16];
            h[0]  = 1.0f;        h[1]  = -a1;
            p1[0] = -a1;         p1[1] = -a1 * p1[0] - a2;   // state y[-1]=1
            p2[0] = -a2;         p2[1] = -a1 * p2[0];        // state y[-2]=1
            #pragma unroll
            for (int i = 2; i < 16; ++i) {
                h[i]  = -a1 * h[i-1]  - a2 * h[i-2];
                p1[i] = -a1 * p1[i-1] - a2 * p1[i-2];
                p2[i] = -a1 * p2[i-1] - a2 * p2[i-2];
            }
            #pragma unroll
            for (int i = 0; i < 16; ++i) {
                hbuf[f][0][i] = h[i];
                hbuf[f][1][i] = p1[i];
                hbuf[f][2][i] = p2[i];
            }
        }
    }

    // ---- hoisted per-lane WMMA A-fragments and state-seed constants ----
    // A 16x4 layout: VGPR0 = K(local) {0|2}, VGPR1 = K(local) {1|3}, m = lane&15.
    v2f A1[4], A2[4];
    float P1a[8], P2a[8], P1b[8], P2b[8];
    #pragma unroll
    for (int k = 0; k < 4; ++k) {
        const int c0 = 4 * k + 2 * hi;        // global K column of A.x
        const int i0 = n - c0;                // h index (H[m][k] = h[m-k])
        const int i1 = i0 - 1;
        const float h0a = hbuf[0][0][i0 < 0 ? 0 : i0];
        const float h1a = hbuf[0][0][i1 < 0 ? 0 : i1];
        const float h0b = hbuf[1][0][i0 < 0 ? 0 : i0];
        const float h1b = hbuf[1][0][i1 < 0 ? 0 : i1];
        A1[k].x = (i0 >= 0) ? h0a : 0.0f;
        A1[k].y = (i1 >= 0) ? h1a : 0.0f;
        A2[k].x = (i0 >= 0) ? h0b : 0.0f;
        A2[k].y = (i1 >= 0) ? h1b : 0.0f;
    }
    #pragma unroll
    for (int v = 0; v < 8; ++v) {
        const int m = v + 8 * hi;             // C/D row held in acc[v]
        P1a[v] = hbuf[0][1][m];  P2a[v] = hbuf[0][2][m];
        P1b[v] = hbuf[1][1][m];  P2b[v] = hbuf[1][2][m];
    }

    // ---- carried state (per channel n; duplicated across lane halves) ----
    float xc0 = 0.0f, xc1 = 0.0f;             // x[t0-2], x[t0-1]
    if (warm > 0) {
        const float* p = xin + (size_t)(chbase + n) * T_LEN + (warm - 2);
        xc0 = p[0]; xc1 = p[1];
    }
    float h1c0 = 0.0f, h1c1 = 0.0f;           // y_hp[t0-2], y_hp[t0-1]
    float l1c0 = 0.0f, l1c1 = 0.0f;           // y_lp[t0-2], y_lp[t0-1]

    const float* xg = xin  + (size_t)(chbase + ch) * T_LEN;
    float*       yg = yout + (size_t)(chbase + ch) * T_LEN;

    for (int t0 = warm; t0 < t_end; t0 += 16) {
        // 1) stage 16 samples x 16 channels into LDS (64B contig per channel)
        {
            const float* p = xg + t0 + halfb * 8;
            v4f xa = *(const v4f*)p;
            v4f xb = *(const v4f*)(p + 4);
            v2f* xp = (v2f*)&xt[ch * TSTR + halfb * 8];
            v2f w;
            w.x = xa.x; w.y = xa.y; xp[0] = w;
            w.x = xa.z; w.y = xa.w; xp[1] = w;
            w.x = xb.x; w.y = xb.y; xp[2] = w;
            w.x = xb.z; w.y = xb.w; xp[3] = w;
        }

        // 2) FILTER 1 (highpass): seed C with state response, 4 chained WMMAs
        v8f acc;
        #pragma unroll
        for (int v = 0; v < 8; ++v) acc[v] = P1a[v] * h1c1 + P2a[v] * h1c0;

        #pragma unroll
        for (int k = 0; k < 4; ++k) {
            const int tau = 4 * k + 2 * hi;                 // B rows tau, tau+1
            const int rb  = n * TSTR;
            v2f lo  = *(const v2f*)&xt[rb + (tau >= 2 ? tau - 2 : 0)];
            v2f hi2 = *(const v2f*)&xt[rb + tau];
            const bool nc = (k == 0) && (hi == 0);          // needs carry halo
            float xm2 = nc ? xc0 : lo.x;
            float xm1 = nc ? xc1 : lo.y;
            v2f B;
            B.x = hp.b0 * hi2.x + hp.b1 * xm1  + hp.b2 * xm2;   // xf[tau]
            B.y = hp.b0 * hi2.y + hp.b1 * hi2.x + hp.b2 * xm1;  // xf[tau+1]
            acc = __builtin_amdgcn_wmma_f32_16x16x4_f32(
                      false, A1[k], false, B, (short)0, acc, false, false);
        }

        // 3) spill highpass block, grab next hp carry (y_hp[t0+14], [t0+15])
        #pragma unroll
        for (int v = 0; v < 8; ++v) y1t[n * TSTR + v + 8 * hi] = acc[v];
        v2f hc = *(const v2f*)&y1t[n * TSTR + 14];

        // 4) FILTER 2 (lowpass) on y_hp, using OLD hp carry for the halo
        v8f acc2;
        #pragma unroll
        for (int v = 0; v < 8; ++v) acc2[v] = P1b[v] * l1c1 + P2b[v] * l1c0;

        #pragma unroll
        for (int k = 0; k < 4; ++k) {
            const int tau = 4 * k + 2 * hi;
            const int rb  = n * TSTR;
            v2f lo  = *(const v2f*)&y1t[rb + (tau >= 2 ? tau - 2 : 0)];
            v2f hi2 = *(const v2f*)&y1t[rb + tau];
            const bool nc = (k == 0) && (hi == 0);
            float ym2 = nc ? h1c0 : lo.x;
            float ym1 = nc ? h1c1 : lo.y;
            v2f B;
            B.x = lp.b0 * hi2.x + lp.b1 * ym1  + lp.b2 * ym2;
            B.y = lp.b0 * hi2.y + lp.b1 * hi2.x + lp.b2 * ym1;
            acc2 = __builtin_amdgcn_wmma_f32_16x16x4_f32(
                       false, A2[k], false, B, (short)0, acc2, false, false);
        }

        // 5) spill lowpass block, update all carried state
        #pragma unroll
        for (int v = 0; v < 8; ++v) y2t[n * TSTR + v + 8 * hi] = acc2[v];
        v2f lc  = *(const v2f*)&y2t[n * TSTR + 14];
        v2f xcn = *(const v2f*)&xt [n * TSTR + 14];
        h1c0 = hc.x;  h1c1 = hc.y;
        l1c0 = lc.x;  l1c1 = lc.y;
        xc0  = xcn.x; xc1  = xcn.y;

        // 6) coalesced 128b stores (skip warm-up region; uniform branch)
        if (t0 >= t_start) {
            const int ob = ch * TSTR + halfb * 8;
            v4f o0, o1;
            o0.x = y2t[ob + 0]; o0.y = y2t[ob + 1];
            o0.z = y2t[ob + 2]; o0.w = y2t[ob + 3];
            o1.x = y2t[ob + 4]; o1.y = y2t[ob + 5];
            o1.z = y2t[ob + 6]; o1.w = y2t[ob + 7];
            float* q = yg + t0 + halfb * 8;
            *(v4f*)q       = o0;
            *(v4f*)(q + 4) = o1;
        }
    }
}

static Coef make_coef(double freq, bool highpass)
{
    const double Q  = 0.7071067811865476;
    const double w0 = 2.0 * M_PI * freq / 16000.0;
    const double cw = cos(w0), sw = sin(w0);
    const double alpha = sw / (2.0 * Q);
    double b0, b1, b2;
    if (highpass) { b0 = (1.0 + cw) / 2.0; b1 = -(1.0 + cw); b2 = (1.0 + cw) / 2.0; }
    else          { b0 = (1.0 - cw) / 2.0; b1 =  (1.0 - cw); b2 = (1.0 - cw) / 2.0; }
    const double a0 = 1.0 + alpha, a1 = -2.0 * cw, a2 = 1.0 - alpha;
    Coef c;
    c.b0 = (float)(b0 / a0); c.b1 = (float)(b1 / a0); c.b2 = (float)(b2 / a0);
    c.a1 = (float)(a1 / a0); c.a2 = (float)(a2 / a0);
    return c;
}

extern "C" void kernel_launch(void* const* d_in, const int* in_sizes, int n_in,
                              void* d_out, int out_size, void* d_ws, size_t ws_size,
                              hipStream_t stream)
{
    (void)in_sizes; (void)n_in; (void)out_size; (void)d_ws; (void)ws_size;
    const float* x = (const float*)d_in[0];
    float*       y = (float*)d_out;
    const Coef hp = make_coef(125.0,  true);
    const Coef lp = make_coef(6000.0, false);
    dim3 grid((N_CH / 16) * NSEG);   // 8 channel-groups x 40 time segments
    dim3 block(32);                  // one wave32 per workgroup
    MicResponseAugment_biquad_wmma<<<grid, block, 0, stream>>>(x, y, hp, lp);
}